// SSAM_35416300323078
// MI455X (gfx1250) — compile-verified
//
#include <hip/hip_runtime.h>
#include <math.h>

typedef float v2f __attribute__((ext_vector_type(2)));
typedef float v8f __attribute__((ext_vector_type(8)));

#define BB 16
#define CC 512
#define HWN 4096
#define KK 10
#define KP 16

// workspace layout (in floats)
#define WB_OFF   0                      // 512*16 = 8192
#define INV_OFF  8192                   // 16
#define PROB_OFF 8208                   // 16*10*4096 = 655360
#define PART_OFF 663568                 // 16*256*16 = 65536
#define RS_OFF   729104                 // 16*16 = 256
// total = 729360 floats = ~2.92 MB

// ---------------------------------------------------------------------------
// Kernel A: pack conv_w into B-matrix layout (c-major, 16 padded k columns)
// and compute 1/||w_k||_2 (padded with 1.0 for k>=10).
// ---------------------------------------------------------------------------
__global__ void ssam_prep(const float* __restrict__ conv_w, float* __restrict__ ws) {
    int t = threadIdx.x;
    float* wB = ws + WB_OFF;
    float* inv_nw = ws + INV_OFF;
    for (int c = t; c < CC; c += 256) {
        #pragma unroll
        for (int k = 0; k < KP; ++k)
            wB[c * KP + k] = (k < KK) ? conv_w[k * CC + c] : 0.0f;
    }
    if (t < KP) {
        float inv = 1.0f;
        if (t < KK) {
            float s = 0.0f;
            for (int c = 0; c < CC; ++c) { float v = conv_w[t * CC + c]; s += v * v; }
            inv = 1.0f / sqrtf(s);
        }
        inv_nw[t] = inv;
    }
}

// ---------------------------------------------------------------------------
// Kernel B: projection GEMM via V_WMMA_F32_16X16X4_F32.
// One wave handles a tile of 16 consecutive spatial positions of one batch.
//   D[pos][k] = sum_c x[b][c][pos] * wB[c][k]
// Then bias, K-norm, scaling, prob store + pooled partial sums.
// ---------------------------------------------------------------------------
__global__ void __launch_bounds__(256) ssam_proj(const float* __restrict__ x,
                                                 const float* __restrict__ conv_b,
                                                 float* __restrict__ ws) {
    const float* wB     = ws + WB_OFF;
    const float* inv_nw = ws + INV_OFF;
    float* prob = ws + PROB_OFF;
    float* part = ws + PART_OFF;

    int wave = threadIdx.x >> 5;
    int lane = threadIdx.x & 31;
    int tile = blockIdx.x * 8 + wave;        // 0..4095
    int b      = tile >> 8;
    int t_in_b = tile & 255;
    int pos0   = t_in_b << 4;
    int half   = lane >> 4;                  // 0: lanes 0-15, 1: lanes 16-31
    int n      = lane & 15;                  // pos index (A) / k column (B,D)

    const float* xcol = x + (size_t)b * CC * HWN + pos0 + n;  // stride HWN per c
    const float* wcol = wB + n;                               // stride KP per c
    int cbase = half * 2;

    v8f acc0 = {};
    v8f acc1 = {};
    #pragma unroll 4
    for (int cc = 0; cc < CC; cc += 8) {
        int c0 = cc + cbase;
        v2f a0 = { xcol[(size_t)c0 * HWN], xcol[(size_t)(c0 + 1) * HWN] };
        v2f b0 = { wcol[c0 * KP],          wcol[(c0 + 1) * KP] };
        acc0 = __builtin_amdgcn_wmma_f32_16x16x4_f32(false, a0, false, b0,
                                                     (short)0, acc0, false, false);
        int c1 = cc + 4 + cbase;
        v2f a1 = { xcol[(size_t)c1 * HWN], xcol[(size_t)(c1 + 1) * HWN] };
        v2f b1 = { wcol[c1 * KP],          wcol[(c1 + 1) * KP] };
        acc1 = __builtin_amdgcn_wmma_f32_16x16x4_f32(false, a1, false, b1,
                                                     (short)0, acc1, false, false);
    }

    float bias = (n < KK) ? conv_b[n] : 0.0f;
    float iw   = inv_nw[n];
    float psum = 0.0f;
    float pv[8];
    #pragma unroll
    for (int j = 0; j < 8; ++j) {
        float v = acc0[j] + acc1[j];
        v = (n < KK) ? (v + bias) : 0.0f;
        // L2 norm over K (16-lane half holds one row of D)
        float s = v * v;
        s += __shfl_xor(s, 1, 32);
        s += __shfl_xor(s, 2, 32);
        s += __shfl_xor(s, 4, 32);
        s += __shfl_xor(s, 8, 32);
        float p = v * iw / sqrtf(s);
        pv[j] = p;
        psum += p;
    }

    // store prob[b][n][pos0 + half*8 + j]  (two b128 stores per active lane)
    if (n < KK) {
        float* pb = prob + ((size_t)(b * KK + n)) * HWN + pos0 + half * 8;
        float4 st0 = make_float4(pv[0], pv[1], pv[2], pv[3]);
        float4 st1 = make_float4(pv[4], pv[5], pv[6], pv[7]);
        *(float4*)(pb)     = st0;
        *(float4*)(pb + 4) = st1;
    }

    // pooled partial: sum over the 16 positions of this tile, per k
    psum += __shfl_xor(psum, 16, 32);
    if (half == 0)
        part[(size_t)(b * 256 + t_in_b) * KP + n] = psum;
}

// ---------------------------------------------------------------------------
// Kernel C: deterministic pooled reduction + SE MLP -> rs[b][k] = gamma*ratio
// ---------------------------------------------------------------------------
__global__ void ssam_se(const float* __restrict__ lin1_w,
                        const float* __restrict__ lin2_w,
                        const float* __restrict__ gamma,
                        float* __restrict__ ws) {
    __shared__ float s_pool[BB][KP];
    __shared__ float s_h[BB][40];
    const float* part = ws + PART_OFF;
    float* rs = ws + RS_OFF;
    int t = threadIdx.x;
    {
        int b = t >> 4, n = t & 15;
        float s = 0.0f;
        for (int i = 0; i < 256; ++i)
            s += part[(size_t)(b * 256 + i) * KP + n];
        s_pool[b][n] = s * (1.0f / 4096.0f);
    }
    __syncthreads();
    for (int idx = t; idx < BB * 40; idx += 256) {
        int b = idx / 40, j = idx % 40;
        float s = 0.0f;
        #pragma unroll
        for (int k = 0; k < KK; ++k) s += s_pool[b][k] * lin1_w[j * KK + k];
        s_h[b][j] = fmaxf(s, 0.0f);
    }
    __syncthreads();
    float g = gamma[0];
    for (int idx = t; idx < BB * KP; idx += 256) {
        int b = idx >> 4, k = idx & 15;
        float r = 0.0f;
        if (k < KK) {
            float s = 0.0f;
            #pragma unroll
            for (int j = 0; j < 40; ++j) s += s_h[b][j] * lin2_w[k * 40 + j];
            r = g / (1.0f + expf(-s));
        }
        rs[idx] = r;
    }
}

// ---------------------------------------------------------------------------
// Kernel D: out[b][c][p] = x[b][c][p] + sum_k prob[b][k][p] * w[k][c] * rs[b][k]
// Block = (b, group of 16 c). prob chunk staged in LDS, float4 streaming.
// ---------------------------------------------------------------------------
__global__ void __launch_bounds__(256) ssam_out(const float* __restrict__ x,
                                                const float* __restrict__ conv_w,
                                                const float* __restrict__ ws,
                                                float* __restrict__ out) {
    __shared__ float sprob[KK * 1024];
    __shared__ float s_w[16][KK];
    const float* prob = ws + PROB_OFF;
    const float* rs   = ws + RS_OFF;
    int t  = threadIdx.x;
    int b  = blockIdx.x >> 5;
    int c0 = (blockIdx.x & 31) << 4;

    if (t < 16 * KK) {
        int ci = t / KK, k = t % KK;
        s_w[ci][k] = conv_w[k * CC + c0 + ci] * rs[b * KP + k];
    }

    for (int chunk = 0; chunk < 4; ++chunk) {
        int p0 = chunk << 10;
        __syncthreads();
        for (int i = t; i < KK * 256; i += 256) {
            int k   = i >> 8;
            int off = (i & 255) << 2;
            *(float4*)&sprob[(k << 10) + off] =
                *(const float4*)&prob[(((size_t)(b * KK + k)) << 12) + p0 + off];
        }
        __syncthreads();
        int po = t << 2;
        for (int ci = 0; ci < 16; ++ci) {
            size_t base = (((size_t)(b * CC + c0 + ci)) << 12) + p0 + po;
            float4 acc = *(const float4*)(x + base);
            #pragma unroll
            for (int k = 0; k < KK; ++k) {
                float w = s_w[ci][k];
                float4 pvv = *(const float4*)&sprob[(k << 10) + po];
                acc.x += pvv.x * w; acc.y += pvv.y * w;
                acc.z += pvv.z * w; acc.w += pvv.w * w;
            }
            *(float4*)(out + base) = acc;
        }
    }
}

// ---------------------------------------------------------------------------
extern "C" void kernel_launch(void* const* d_in, const int* in_sizes, int n_in,
                              void* d_out, int out_size, void* d_ws, size_t ws_size,
                              hipStream_t stream) {
    const float* x      = (const float*)d_in[0];
    const float* conv_w = (const float*)d_in[1];
    const float* conv_b = (const float*)d_in[2];
    const float* lin1_w = (const float*)d_in[3];
    const float* lin2_w = (const float*)d_in[4];
    const float* gamma  = (const float*)d_in[5];
    float* out = (float*)d_out;
    float* ws  = (float*)d_ws;

    ssam_prep<<<1, 256, 0, stream>>>(conv_w, ws);
    ssam_proj<<<512, 256, 0, stream>>>(x, conv_b, ws);
    ssam_se<<<1, 256, 0, stream>>>(lin1_w, lin2_w, gamma, ws);
    ssam_out<<<512, 256, 0, stream>>>(x, conv_w, ws, out);
}